// DLA_85066122265073
// MI455X (gfx1250) — compile-verified
//
#include <hip/hip_runtime.h>

// Problem constants (from reference setup_inputs)
#define BD   4
#define REF  128
#define SD   512
#define DD   64
#define ET   128
#define NH   128
#define HH   4
#define EK   32
#define VCHUNK  64                 // value rows per TDM chunk (16KB)
#define NCHUNK  (SD / VCHUNK)      // 8 chunks, double-buffered

typedef float v2f __attribute__((ext_vector_type(2)));
typedef float v8f __attribute__((ext_vector_type(8)));
typedef unsigned u32x4 __attribute__((ext_vector_type(4)));
typedef unsigned u32x8 __attribute__((ext_vector_type(8)));

// ---------------------------------------------------------------------------
// Tensor Data Mover: async 1-D copy of VCHUNK*DD f32 (16KB) global -> LDS.
// Hand-packed D# (CDNA5 ISA): group0 = {count=1, lds_addr,
// global_addr[56:0], type=2}; group1 = {data_size=4B, tensor_dim0 = tile_dim0
// = stride0 = 4096 elems, 1-D tile, no pad/iterate, workgroup_mask=0}.
// VADDR2/3 omitted (<=2D tensor). Tracked by TENSORcnt (in-order per wave).
// ---------------------------------------------------------------------------
__device__ __forceinline__ void dla_tdm_load_16k(unsigned lds_off,
                                                 const void* gptr) {
  const unsigned long long ga = (unsigned long long)gptr;
  const unsigned NE = VCHUNK * DD;            // 4096 elements
  u32x4 g0;
  g0[0] = 1u;                                  // count=1 (valid user D#)
  g0[1] = lds_off;                             // LDS byte address
  g0[2] = (unsigned)ga;                        // global_addr[31:0]
  g0[3] = (unsigned)((ga >> 32) & 0x1FFFFFFu)  // global_addr[56:32]
          | (2u << 30);                        // type = 2 ("image")
  u32x8 g1;
  g1[0] = 2u << 16;                            // data_size: 2 -> 4 bytes
  g1[1] = (NE & 0xFFFFu) << 16;                // tensor_dim0[15:0] @ bits 63:48
  g1[2] = (NE >> 16) | (1u << 16);             // tensor_dim0[31:16] | tensor_dim1=1
  g1[3] = (NE & 0xFFFFu) << 16;                // tile_dim0 @ bits 127:112
  g1[4] = 0u;                                  // tile_dim1 = tile_dim2 = 0 (1-D)
  g1[5] = NE;                                  // tensor_dim0_stride[31:0]
  g1[6] = 0u;
  g1[7] = 0u;
  asm volatile("tensor_load_to_lds %0, %1" :: "s"(g0), "s"(g1) : "memory");
}

// ---------------------------------------------------------------------------
// C = scale * (A @ W^T) + bias using native fp32 WMMA (v_wmma_f32_16x16x4_f32)
// A: (M x K) row-major, lda;  W: (N x K) row-major, ldw  (B-operand = W^T).
// One wave per 16x32 output strip: A-fragment reused by two B-fragments
// (2 WMMAs per k-step). grid.z slices (A,W,C) for the batched scores GEMM.
//
// f32 16x16x4 VGPR layout (ISA 7.12.2):
//   A 16x4: lanes 0-15 row M hold K{0,1}; lanes 16-31 hold K{2,3}
//   B 4x16: lanes 0-15 col N hold K{0,1}; lanes 16-31 hold K{2,3}
//   C/D   : VGPR i -> row (i + 8*half), col = lane&15
//
// OUTMODE 0: C row-major (ldc).  OUTMODE 1: head-split scatter
//   C[((b*H+h)*RPB + r)*EK + e],  b = m/RPB, h = n/EK  (RPB constexpr pow2).
// ---------------------------------------------------------------------------
template <int K, int OUTMODE, int RPB>
__global__ void dla_gemm_f32_wmma(const float* __restrict__ A,
                                  const float* __restrict__ W,
                                  const float* __restrict__ bias,
                                  float* __restrict__ C,
                                  int lda, int ldw, int ldc, float scale,
                                  long aslice, long wslice, long cslice) {
  const int lane = threadIdx.x & 31;
  const int m0 = blockIdx.x * 16;
  const int n0 = blockIdx.y * 32;
  const long z = blockIdx.z;
  A += z * aslice;
  W += z * wslice;
  C += z * cslice;

  const int half = lane >> 4;   // which K-pair this lane holds
  const int row  = lane & 15;   // M row for A, N col for B/C

  const float* ar = A + (long)(m0 + row) * lda + half * 2;
  const float* w0 = W + (long)(n0 + row) * ldw + half * 2;
  const float* w1 = w0 + (long)16 * ldw;

  v8f acc0 = {}, acc1 = {};
#pragma unroll 8
  for (int k0 = 0; k0 < K; k0 += 4) {
    v2f af, b0, b1;
    af[0] = ar[k0]; af[1] = ar[k0 + 1];
    b0[0] = w0[k0]; b0[1] = w0[k0 + 1];
    b1[0] = w1[k0]; b1[1] = w1[k0 + 1];
    acc0 = __builtin_amdgcn_wmma_f32_16x16x4_f32(
        false, af, false, b0, (short)0, acc0, false, false);
    acc1 = __builtin_amdgcn_wmma_f32_16x16x4_f32(
        false, af, false, b1, (short)0, acc1, false, false);
  }

#pragma unroll
  for (int t = 0; t < 2; ++t) {
    const v8f& acc = t ? acc1 : acc0;
    const int n = n0 + t * 16 + row;
    const float bv = bias ? bias[n] : 0.0f;
#pragma unroll
    for (int i = 0; i < 8; ++i) {
      const int m = m0 + i + half * 8;
      const float v = acc[i] * scale + bv;
      if (OUTMODE == 0) {
        C[(long)m * ldc + n] = v;
      } else {
        const int bb = m / RPB, rr = m % RPB;        // RPB constexpr pow2
        const int hh = n >> 5, ee = n & (EK - 1);    // EK == 32
        C[(((long)(bb * HH + hh) * RPB + rr) * EK) + ee] = v;
      }
    }
  }
}

// ---------------------------------------------------------------------------
// stride = sigmoid(stride_in @ Wr^T + br). One block per b, D threads.
// ---------------------------------------------------------------------------
__global__ void dla_stride_kernel(const float* __restrict__ stride_in,
                                  const float* __restrict__ Wr,
                                  const float* __restrict__ br,
                                  float* __restrict__ ws_stride,
                                  float* __restrict__ out_stride) {
  const int b = blockIdx.x;
  const int d = threadIdx.x;
  const float* si = stride_in + b * DD;
  float acc = br[d];
#pragma unroll 8
  for (int c = 0; c < DD; ++c) acc = fmaf(si[c], Wr[d * DD + c], acc);
  const float s = 1.0f / (1.0f + __expf(-acc));
  ws_stride[b * DD + d]  = s;
  out_stride[b * DD + d] = s;
}

// ---------------------------------------------------------------------------
// Pack mask(b,s,d) (int 0/1, d=0..63) into two 32-bit words per (b,s).
// ---------------------------------------------------------------------------
__global__ void dla_pack_mask(const int* __restrict__ mask,
                              unsigned* __restrict__ mpack) {
  const int idx = blockIdx.x * 256 + threadIdx.x;   // b*S + s, 2048 total
  const int* mp = mask + (long)idx * DD;
  unsigned lo = 0u, hi = 0u;
#pragma unroll
  for (int j = 0; j < 32; ++j) {
    lo |= (mp[j]      != 0 ? 1u : 0u) << j;
    hi |= (mp[32 + j] != 0 ? 1u : 0u) << j;
  }
  mpack[2 * idx]     = lo;
  mpack[2 * idx + 1] = hi;
}

// ---------------------------------------------------------------------------
// Fused per-d masked softmax over S + weighted sum with value.
// Block = (b, r); 256 threads = (h, d) pairs.
//   - window test |tt - qt| <= stride(b,d): |tt-qt| precomputed once in LDS
//   - packed mask + 4 score rows in LDS  -> pass 1 (max) touches no global mem
//   - value streamed via TDM (tensor_load_to_lds), double-buffered 16KB
//     chunks: DMA of chunk c+1 overlaps compute on chunk c; the first two
//     DMAs overlap the entire max pass. Retired with s_wait_tensorcnt.
// Masked entries are exactly 0 after exp underflow (reference uses -1e9 fill);
// all-masked rows degenerate to a uniform softmax -> mean(value) fallback.
// ---------------------------------------------------------------------------
__global__ void dla_attn_softmax(const float* __restrict__ scores,
                                 const float* __restrict__ value,
                                 const unsigned* __restrict__ mpack,
                                 const float* __restrict__ tt,
                                 const float* __restrict__ qt,
                                 const float* __restrict__ stridev,
                                 float* __restrict__ x) {
  __shared__ float    s_ad[SD];               // |tt - qt|
  __shared__ float    s_sc[HH][SD];           // score rows for the 4 heads
  __shared__ unsigned s_mlo[SD], s_mhi[SD];   // packed mask bits
  __shared__ __align__(16) float s_val[2][VCHUNK * DD];   // 2 x 16KB

  const int b = blockIdx.x;
  const int r = blockIdx.y;
  const int tid = threadIdx.x;                // 0..255
  const float qv = qt[r];
  const float* vbase = value + (long)b * SD * DD;

  // Kick off the first two value chunks on the TDM engine; they run while
  // we stage the control data and run the (LDS-only) max pass.
  if (tid == 0) {
    dla_tdm_load_16k((unsigned)(size_t)&s_val[0][0], vbase);
    dla_tdm_load_16k((unsigned)(size_t)&s_val[1][0], vbase + VCHUNK * DD);
  }

  for (int i = tid; i < SD; i += 256) {
    s_ad[i]  = fabsf(tt[b * SD + i] - qv);
    s_mlo[i] = mpack[2 * (b * SD + i)];
    s_mhi[i] = mpack[2 * (b * SD + i) + 1];
  }
#pragma unroll
  for (int hh = 0; hh < HH; ++hh)
    for (int i = tid; i < SD; i += 256)
      s_sc[hh][i] = scores[(((long)(b * HH + hh) * REF) + r) * SD + i];
  __syncthreads();

  const int h = tid >> 6;                     // 0..3
  const int d = tid & (DD - 1);               // 0..63
  const float sd = stridev[b * DD + d];
  const unsigned* mrow = (d < 32) ? s_mlo : s_mhi;
  const int shift = d & 31;

  // pass 1: row max over unmasked entries (LDS only; overlaps the TDM DMAs)
  float m = -1.0e9f;
  for (int s = 0; s < SD; ++s) {
    const bool am = (s_ad[s] <= sd) & (((mrow[s] >> shift) & 1u) != 0u);
    if (am) m = fmaxf(m, s_sc[h][s]);
  }

  // pass 2: numerator / denominator, value streamed through LDS by the TDM
  float num = 0.0f, den = 0.0f, sv = 0.0f;
  for (int c = 0; c < NCHUNK; ++c) {
    if (tid == 0) {
      // SIMM16 of s_wait_tensorcnt must be an immediate: two constant sites.
      if (c < NCHUNK - 1) __builtin_amdgcn_s_wait_tensorcnt(1);  // chunk c done
      else                __builtin_amdgcn_s_wait_tensorcnt(0);  // last chunk
    }
    __syncthreads();
    const float* vb = &s_val[c & 1][0];
#pragma unroll 4
    for (int s = c * VCHUNK; s < (c + 1) * VCHUNK; ++s) {
      const float vv = vb[(s - c * VCHUNK) * DD + d];
      sv += vv;
      const bool am = (s_ad[s] <= sd) & (((mrow[s] >> shift) & 1u) != 0u);
      if (am) {
        const float e = __expf(s_sc[h][s] - m);
        num = fmaf(e, vv, num);
        den += e;
      }
    }
    __syncthreads();                           // everyone done reading buf[c&1]
    if (tid == 0 && c + 2 < NCHUNK)
      dla_tdm_load_16k((unsigned)(size_t)&s_val[c & 1][0],
                       vbase + (c + 2) * VCHUNK * DD);
  }
  const float outv = (den > 0.0f) ? (num / den) : (sv * (1.0f / (float)SD));
  // x layout: (B, REF, H*D), col = h*D + d  (matches transpose+reshape)
  x[((long)(b * REF + r)) * (HH * DD) + h * DD + d] = outv;
}

// ---------------------------------------------------------------------------
extern "C" void kernel_launch(void* const* d_in, const int* in_sizes, int n_in,
                              void* d_out, int out_size, void* d_ws, size_t ws_size,
                              hipStream_t stream) {
  const float* query     = (const float*)d_in[0];   // (B,REF,ET)
  const float* key_t     = (const float*)d_in[1];   // (B,S,ET)
  const float* value     = (const float*)d_in[2];   // (B,S,D)
  const int*   mask      = (const int*)  d_in[3];   // (B,S,D)
  const float* qt        = (const float*)d_in[4];   // (REF,)
  const float* tt        = (const float*)d_in[5];   // (B,S)
  const float* stride_in = (const float*)d_in[6];   // (B,1,1,D)
  const float* Wq        = (const float*)d_in[7];   // (ET,ET)
  const float* bq        = (const float*)d_in[8];
  const float* Wk        = (const float*)d_in[9];
  const float* bk        = (const float*)d_in[10];
  const float* Wo        = (const float*)d_in[11];  // (NH, H*D)
  const float* bo        = (const float*)d_in[12];
  const float* Wr        = (const float*)d_in[13];  // (D,D)
  const float* br        = (const float*)d_in[14];

  float* out = (float*)d_out;            // [0,65536) = out, [65536,65792) = stride
  float* ws  = (float*)d_ws;

  float*    qproj  = ws;                 // (B,H,REF,EK)   65536 f32
  float*    kproj  = qproj + 65536;      // (B,H,S,EK)    262144 f32
  float*    scores = kproj + 262144;     // (B,H,REF,S)  1048576 f32
  float*    x      = scores + 1048576;   // (B,REF,H*D)   131072 f32
  float*    strv   = x + 131072;         // (B,D)            256 f32
  unsigned* mpack  = (unsigned*)(strv + 256);  // (B,S,2)    4096 u32

  // 1) stride gate (also output #2)
  dla_stride_kernel<<<dim3(BD), dim3(DD), 0, stream>>>(
      stride_in, Wr, br, strv, out + (long)BD * REF * NH);

  // 2) pack the 0/1 mask into 64 bits per (b,s)
  dla_pack_mask<<<dim3(BD * SD / 256), dim3(256), 0, stream>>>(mask, mpack);

  // 3) Q projection: (512x128)@(128x128)^T + bq -> head-split (b,h,r,e)
  dla_gemm_f32_wmma<ET, 1, REF><<<dim3(32, 4, 1), dim3(32), 0, stream>>>(
      query, Wq, bq, qproj, ET, ET, 0, 1.0f, 0, 0, 0);

  // 4) K projection: (2048x128)@(128x128)^T + bk -> head-split (b,h,s,e)
  dla_gemm_f32_wmma<ET, 1, SD><<<dim3(128, 4, 1), dim3(32), 0, stream>>>(
      key_t, Wk, bk, kproj, ET, ET, 0, 1.0f, 0, 0, 0);

  // 5) scores(b,h,r,s) = q·k / sqrt(32); batched over z = b*H+h (16 slices)
  dla_gemm_f32_wmma<EK, 0, 1><<<dim3(REF / 16, SD / 32, BD * HH), dim3(32), 0, stream>>>(
      qproj, kproj, nullptr, scores, EK, EK, SD, 0.17677669529663687f,
      (long)REF * EK, (long)SD * EK, (long)REF * SD);

  // 6) fused window-mask + per-d softmax over S + p@value (TDM-fed)
  dla_attn_softmax<<<dim3(BD, REF), dim3(256), 0, stream>>>(
      scores, value, mpack, tt, qt, strv, x);

  // 7) out = x @ Wo^T + bo : (512x256)@(256x128)^T
  dla_gemm_f32_wmma<HH * DD, 0, 1><<<dim3(32, 4, 1), dim3(32), 0, stream>>>(
      x, Wo, bo, out, HH * DD, HH * DD, NH, 1.0f, 0, 0, 0);
}